// ndcg_87084756894039
// MI455X (gfx1250) — compile-verified
//
#include <hip/hip_runtime.h>
#include <hip/hip_bf16.h>
#include <math.h>

// ---------------- problem constants ----------------
#define QROWS   1024
#define DDIM    128
#define NCAND   500000
#define TOPK    50
#define QTILE   32                         // queries per block (2 WMMA A-tiles)
#define CHUNK   2048                       // candidates per block (pass 1)
#define NCHUNK  ((NCAND + CHUNK - 1) / CHUNK)   // 245
#define ENTRIES (NCHUNK * TOPK)            // 12250 partial entries per query
#define NB      8                          // B-fragment pipeline depth

typedef float v2f __attribute__((ext_vector_type(2)));
typedef float v8f __attribute__((ext_vector_type(8)));

// =====================================================================
// Pass 1: fp32 WMMA scores for a 32-query x 2048-candidate tile, then
// per-chunk top-50 per query via wave argmax rounds over LDS.
// Two A-tiles per wave -> each B load feeds TWO v_wmma ops (2:1 ratio).
// Explicit 8-deep rotating B buffer forces ~8 loads in flight.
// gridDim = (QROWS/32, NCHUNK), blockDim = 256 (8 waves)
// =====================================================================
__global__ __launch_bounds__(256)
void topk_score_kernel(const float* __restrict__ qry,
                       const float* __restrict__ cand,
                       float* __restrict__ wsv,
                       int*   __restrict__ wsi)
{
    extern __shared__ float scores[];          // QTILE * CHUNK floats = 256 KB
    const int tid   = threadIdx.x;
    const int wave  = tid >> 5;
    const int lane  = tid & 31;
    const int n0    = lane & 15;               // column within 16-wide tile
    const int hi    = lane >> 4;               // K-half selector for f32 WMMA
    const int qTile = blockIdx.x;              // 0..31
    const int chunk = blockIdx.y;              // 0..NCHUNK-1

    // ---- preload two A tiles (query rows) into registers ----
    // fp32 WMMA 16x16x4 A layout: lane L holds row m=L%16, K = {2*hi, 2*hi+1}
    // per 4-wide K step -> columns 4k + 2*hi + {0,1}
    v2f a0[32], a1[32];
    {
        const float* q0 = qry + (size_t)(qTile * QTILE + n0)      * DDIM + 2 * hi;
        const float* q1 = qry + (size_t)(qTile * QTILE + 16 + n0) * DDIM + 2 * hi;
        #pragma unroll
        for (int k = 0; k < 32; ++k) {
            a0[k] = *(const v2f*)(q0 + 4 * k);
            a1[k] = *(const v2f*)(q1 + 4 * k);
        }
    }

    // ---- score 128 sub-tiles of 16 candidates (16 per wave) ----
    for (int t = wave; t < CHUNK / 16; t += 8) {
        const int crow  = chunk * CHUNK + t * 16 + n0;   // candidate row (B col)
        const bool valid = (crow < NCAND);
        const float* cptr = cand + (size_t)(valid ? crow : 0) * DDIM + 2 * hi;

        // prefetch the next sub-tile this wave will touch
        if (t + 8 < CHUNK / 16)
            __builtin_prefetch(cand + (size_t)(chunk * CHUNK + (t + 8) * 16 + n0) * DDIM, 0, 0);

        // explicit rotating B buffer: keeps NB loads in flight by construction
        v2f bb[NB];
        #pragma unroll
        for (int k = 0; k < NB; ++k)
            bb[k] = *(const v2f*)(cptr + 4 * k);

        v8f acc0 = {};
        v8f acc1 = {};
        #pragma unroll
        for (int k = 0; k < 32; ++k) {
            v2f b = bb[k & (NB - 1)];
            if (k + NB < 32)                       // refill slot NB steps ahead
                bb[k & (NB - 1)] = *(const v2f*)(cptr + 4 * (k + NB));
            // one B fragment feeds two independent fp32 WMMA chains
            acc0 = __builtin_amdgcn_wmma_f32_16x16x4_f32(
                       false, a0[k], false, b, (short)0, acc0, false, false);
            acc1 = __builtin_amdgcn_wmma_f32_16x16x4_f32(
                       false, a1[k], false, b, (short)0, acc1, false, false);
        }

        // C layout: VGPR r -> row (r + 8*hi), column n0 of this tile
        #pragma unroll
        for (int r = 0; r < 8; ++r) {
            float v0 = valid ? acc0[r] : -__builtin_inff();
            float v1 = valid ? acc1[r] : -__builtin_inff();
            scores[(r + 8 * hi)      * CHUNK + t * 16 + n0] = v0;
            scores[(16 + r + 8 * hi) * CHUNK + t * 16 + n0] = v1;
        }
    }
    __syncthreads();

    // ---- per-chunk top-50: each wave owns 4 query rows ----
    for (int qq = 0; qq < 4; ++qq) {
        const int mq = wave * 4 + qq;                 // 0..31
        float* row = scores + mq * CHUNK;
        for (int j = 0; j < TOPK; ++j) {
            float best = -__builtin_inff();
            int   bi   = -1;
            for (int i = lane; i < CHUNK; i += 32) {
                float v = row[i];
                if (v > best) { best = v; bi = i; }
            }
            // butterfly argmax: every lane ends with the winner
            #pragma unroll
            for (int off = 16; off > 0; off >>= 1) {
                float ov = __shfl_xor(best, off, 32);
                int   oi = __shfl_xor(bi,   off, 32);
                if (ov > best || (ov == best && oi != -1 && (bi == -1 || oi < bi))) {
                    best = ov; bi = oi;
                }
            }
            if (lane == 0) {
                const size_t base =
                    ((size_t)(qTile * QTILE + mq) * NCHUNK + chunk) * TOPK + j;
                wsv[base] = best;
                wsi[base] = (bi >= 0) ? (chunk * CHUNK + bi) : -1;
                if (bi >= 0) row[bi] = -__builtin_inff();   // knock out winner
            }
            // same-wave DS ops are in order; the store aliases the loads
            // above, so the compiler cannot hoist next-round loads over it.
        }
    }
}

// =====================================================================
// Pass 2: merge NCHUNK partial top-50 lists per query -> final top-50.
// gridDim = QROWS, blockDim = 256. Tie-break on smaller candidate index
// to match jax.lax.top_k ordering.
// =====================================================================
__global__ __launch_bounds__(256)
void topk_merge_kernel(const float* __restrict__ wsv,
                       const int*   __restrict__ wsi,
                       float* __restrict__ outv,
                       int*   __restrict__ outi)
{
    extern __shared__ char smem[];
    float* vals = (float*)smem;                 // ENTRIES floats
    int*   idxs = (int*)(vals + ENTRIES);       // ENTRIES ints
    __shared__ float redv[8];
    __shared__ int   redi[8];
    __shared__ int   redp[8];

    const int q    = blockIdx.x;
    const int tid  = threadIdx.x;
    const int lane = tid & 31;
    const int wave = tid >> 5;

    for (int i = tid; i < ENTRIES; i += 256) {
        vals[i] = wsv[(size_t)q * ENTRIES + i];
        idxs[i] = wsi[(size_t)q * ENTRIES + i];
    }
    __syncthreads();

    for (int j = 0; j < TOPK; ++j) {
        float best = -__builtin_inff();
        int   bci  = 0x7fffffff;   // candidate index (tie-break key)
        int   bp   = -1;           // position in LDS
        for (int i = tid; i < ENTRIES; i += 256) {
            float v  = vals[i];
            int   ci = idxs[i];
            if (v > best || (v == best && ci >= 0 && ci < bci)) {
                best = v; bci = ci; bp = i;
            }
        }
        #pragma unroll
        for (int off = 16; off > 0; off >>= 1) {
            float ov = __shfl_xor(best, off, 32);
            int   oc = __shfl_xor(bci,  off, 32);
            int   op = __shfl_xor(bp,   off, 32);
            if (ov > best || (ov == best && oc < bci)) {
                best = ov; bci = oc; bp = op;
            }
        }
        if (lane == 0) { redv[wave] = best; redi[wave] = bci; redp[wave] = bp; }
        __syncthreads();
        if (tid == 0) {
            float fb = redv[0]; int fc = redi[0]; int fp = redp[0];
            #pragma unroll
            for (int w = 1; w < 8; ++w) {
                if (redv[w] > fb || (redv[w] == fb && redi[w] < fc)) {
                    fb = redv[w]; fc = redi[w]; fp = redp[w];
                }
            }
            outv[(size_t)q * TOPK + j] = fb;
            outi[(size_t)q * TOPK + j] = (fc == 0x7fffffff) ? -1 : fc;
            if (fp >= 0) vals[fp] = -__builtin_inff();
        }
        __syncthreads();
    }
}

// =====================================================================
extern "C" void kernel_launch(void* const* d_in, const int* in_sizes, int n_in,
                              void* d_out, int out_size, void* d_ws, size_t ws_size,
                              hipStream_t stream)
{
    const float* qry  = (const float*)d_in[0];   // [1024,128] f32
    const float* cand = (const float*)d_in[1];   // [500000,128] f32
    // d_in[2] = k (always 50 here)

    // workspace: per-(query,chunk) top-50 values then indices
    float* wsv = (float*)d_ws;
    int*   wsi = (int*)(wsv + (size_t)QROWS * NCHUNK * TOPK);

    // outputs: values [1024,50] f32, then indices [1024,50] i32
    float* outv = (float*)d_out;
    int*   outi = (int*)((float*)d_out + (size_t)QROWS * TOPK);

    dim3 grid1(QROWS / QTILE, NCHUNK);           // qTile fastest-varying:
    size_t shmem1 = (size_t)QTILE * CHUNK * sizeof(float);   // 256 KB LDS
    topk_score_kernel<<<grid1, 256, shmem1, stream>>>(qry, cand, wsv, wsi);

    size_t shmem2 = (size_t)ENTRIES * (sizeof(float) + sizeof(int)); // ~98 KB
    topk_merge_kernel<<<QROWS, 256, shmem2, stream>>>(wsv, wsi, outv, outi);
}